// BatchNorm_33363305956121
// MI455X (gfx1250) — compile-verified
//
#include <hip/hip_runtime.h>

// Segment-norm for MI455X (gfx1250, wave32).
// One 256-thread block (8 wave32) per segment; batch is sorted so segments are
// contiguous ranges found via (scalarized) binary search. Pass 1 streams the
// segment through LDS with double-buffered CDNA5 async global->LDS B128 copies
// (ASYNCcnt / s_wait_asynccnt), computes sum/sumsq/|v|^2 with b128 LDS reads.
// Pass 2 applies the normalization with b128 non-temporal global loads/stores.

typedef float v4f __attribute__((ext_vector_type(4)));

#define SDIM 256
#define VDIM 64
#define VROW 192          // VDIM * 3
#define NSEG 512
#define CH   8            // rows per LDS chunk
#define EPSF 1e-6f

__device__ __forceinline__ void async_copy_b128(unsigned lds_off, const float* gaddr) {
  // CDNA5 async DMA: copies 16 bytes per enabled lane from global memory to the LDS offset in VDST.
  asm volatile("global_load_async_to_lds_b128 %0, %1, off"
               :: "v"(lds_off), "v"(gaddr) : "memory");
}
__device__ __forceinline__ void wait_async_le4() {
  asm volatile("s_wait_asynccnt 0x4" ::: "memory");
}
__device__ __forceinline__ void wait_async_0() {
  asm volatile("s_wait_asynccnt 0x0" ::: "memory");
}

__device__ __forceinline__ int lower_bound_i(const int* __restrict__ a, int n, int key) {
  int lo = 0, hi = n;
  while (lo < hi) {
    int mid = (lo + hi) >> 1;
    if (a[mid] < key) lo = mid + 1; else hi = mid;
  }
  return lo;
}

__device__ __forceinline__ float rcp_sqrt_clamped(float var) {
  return 1.0f / sqrtf(fmaxf(fmaxf(var, 0.0f), EPSF));
}

__global__ __launch_bounds__(256) void seg_norm_kernel(
    const float* __restrict__ sg, const float* __restrict__ vg,
    const int* __restrict__ batch, const float* __restrict__ weight,
    const float* __restrict__ bias, float* __restrict__ dout, int n) {

  __shared__ float sbufS[2][CH][SDIM];   // 16 KB  (async landing pad, s rows)
  __shared__ float sbufV[2][CH][VROW];   // 12 KB  (async landing pad, v rows)
  __shared__ v4f   redSsum[4][64];       // 4 KB   (row-lane partial sums)
  __shared__ v4f   redSsq [4][64];       // 4 KB
  __shared__ v4f   redV   [4][48];       // 3 KB   (reused as compsum[192])
  __shared__ float vinv_sh[VDIM];

  const int b   = blockIdx.x;
  const int tid = threadIdx.x;

  // Segment range (uniform per block; compiler scalarizes this to s_loads).
  const int start = lower_bound_i(batch, n, b);
  const int end   = lower_bound_i(batch, n, b + 1);
  const int rows  = end - start;
  const float cnt     = (float)(rows > 0 ? rows : 1);
  const float inv_cnt = 1.0f / cnt;

  // ---- Async-issue roles: each block-wide issue loads a PAIR of rows.
  //   lanes 0..63   -> 16B slice of the s row (64*16B = 1024B)
  //   lanes 64..111 -> 16B slice of the v row (48*16B =  768B)
  //   lanes 112..127 idle (those waves run with partial EXEC, still 1 issue)
  // => every wave issues exactly CH/2 = 4 async b128 per chunk, so
  //    s_wait_asynccnt 4 pipelines chunk c+1's DMA under chunk c's compute.
  const int  half = tid >> 7;
  const int  lane = tid & 127;
  const bool isS  = lane < 64;
  const bool isV  = (lane >= 64) && (lane < 112);
  const int  sCol = lane << 2;
  const int  vCol = (lane - 64) << 2;

  // ---- Compute roles: thread owns 4 consecutive dims, 4 rows in flight.
  const int  u_s  = tid & 63;      // s dim group: dims [4*u_s, 4*u_s+3]
  const int  rl_s = tid >> 6;      // row-lane 0..3
  const int  u_v  = tid % 48;      // v comp group: comps [4*u_v, 4*u_v+3]
  const int  rl_v = tid / 48;      // row-lane 0..3 (tid < 192)
  const bool hasV = tid < 192;

  v4f sum4 = {0.f, 0.f, 0.f, 0.f};
  v4f sq4  = {0.f, 0.f, 0.f, 0.f};
  v4f vsq4 = {0.f, 0.f, 0.f, 0.f};

  const int nchunk = (rows + CH - 1) / CH;
  int buf = 0;

  auto issue = [&](int bf, int gr0) {
#pragma unroll
    for (int pr = 0; pr < CH / 2; ++pr) {
      const int rl = pr * 2 + half;
      int g = gr0 + rl;
      if (g > end - 1) g = end - 1;   // clamp tail: duplicate row, uniform count
      if (isS) {
        unsigned lo = (unsigned)(unsigned long long)&sbufS[bf][rl][sCol];
        async_copy_b128(lo, sg + (size_t)g * SDIM + sCol);
      } else if (isV) {
        unsigned lo = (unsigned)(unsigned long long)&sbufV[bf][rl][vCol];
        async_copy_b128(lo, vg + (size_t)g * VROW + vCol);
      }
    }
  };

  // ---- Pass 1: double-buffered async stream, accumulate sum / sumsq / |v|^2.
  if (nchunk > 0) issue(0, start);
  for (int c = 0; c < nchunk; ++c) {
    const int gr0 = start + c * CH;
    if (c + 1 < nchunk) { issue(buf ^ 1, gr0 + CH); wait_async_le4(); }
    else                { wait_async_0(); }
    __syncthreads();

    const int cr = (end - gr0 < CH) ? (end - gr0) : CH;
    for (int lr = rl_s; lr < cr; lr += 4) {            // ds_load_b128
      const v4f x = *(const v4f*)&sbufS[buf][lr][4 * u_s];
      sum4 += x;
      sq4  += x * x;
    }
    if (hasV) {
      for (int lr = rl_v; lr < cr; lr += 4) {          // ds_load_b128
        const v4f x = *(const v4f*)&sbufV[buf][lr][4 * u_v];
        vsq4 += x * x;
      }
    }
    __syncthreads();
    buf ^= 1;
  }

  // ---- Cross-row-lane reduction (4 partials per dim group).
  redSsum[rl_s][u_s] = sum4;
  redSsq [rl_s][u_s] = sq4;
  if (hasV) redV[rl_v][u_v] = vsq4;
  __syncthreads();

  const v4f tsum = redSsum[0][u_s] + redSsum[1][u_s] + redSsum[2][u_s] + redSsum[3][u_s];
  const v4f tsq  = redSsq [0][u_s] + redSsq [1][u_s] + redSsq [2][u_s] + redSsq [3][u_s];

  if (tid < 48) {   // per-component totals -> reuse redV[0] as compsum[192]
    const v4f tot = redV[0][tid] + redV[1][tid] + redV[2][tid] + redV[3][tid];
    redV[0][tid] = tot;
  }
  __syncthreads();

  if (tid < VDIM) { // channel sums: 3 consecutive components per channel
    const float* cs = (const float*)&redV[0][0];
    const float ch = cs[3 * tid] + cs[3 * tid + 1] + cs[3 * tid + 2];
    vinv_sh[tid] = 1.0f / fmaxf(ch * inv_cnt, EPSF);
  }

  // ---- s stats, register-resident per thread (4 dims).
  const v4f mean4 = tsum * inv_cnt;
  v4f var4 = tsq * inv_cnt - mean4 * mean4;            // E[x^2] - mean^2
  v4f prec4;
  prec4.x = rcp_sqrt_clamped(var4.x);
  prec4.y = rcp_sqrt_clamped(var4.y);
  prec4.z = rcp_sqrt_clamped(var4.z);
  prec4.w = rcp_sqrt_clamped(var4.w);
  const v4f w4 = *(const v4f*)(weight + 4 * u_s);
  const v4f b4 = *(const v4f*)(bias   + 4 * u_s);
  const v4f scale4 = prec4 * w4;
  const v4f shift4 = b4 - mean4 * scale4;              // out = x*scale + shift
  __syncthreads();

  v4f vscale4 = {0.f, 0.f, 0.f, 0.f};
  if (hasV) {
    vscale4.x = vinv_sh[(4 * u_v + 0) / 3];
    vscale4.y = vinv_sh[(4 * u_v + 1) / 3];
    vscale4.z = vinv_sh[(4 * u_v + 2) / 3];
    vscale4.w = vinv_sh[(4 * u_v + 3) / 3];
  }

  float* __restrict__ soutp = dout;                        // [N, SDIM]
  float* __restrict__ voutp = dout + (size_t)n * SDIM;     // [N, VDIM, 3]

  // ---- Pass 2: apply. b128 NT loads (last use) + b128 NT stores.
  for (int r = start + rl_s; r < end; r += 4) {
    const size_t off = (size_t)r * SDIM + 4 * u_s;
    const v4f x = __builtin_nontemporal_load((const v4f*)(sg + off));
    __builtin_nontemporal_store(x * scale4 + shift4, (v4f*)(soutp + off));
  }
  if (hasV) {
    for (int r = start + rl_v; r < end; r += 4) {
      const size_t off = (size_t)r * VROW + 4 * u_v;
      const v4f x = __builtin_nontemporal_load((const v4f*)(vg + off));
      __builtin_nontemporal_store(x * vscale4, (v4f*)(voutp + off));
    }
  }
}

extern "C" void kernel_launch(void* const* d_in, const int* in_sizes, int n_in,
                              void* d_out, int out_size, void* d_ws, size_t ws_size,
                              hipStream_t stream) {
  (void)n_in; (void)out_size; (void)d_ws; (void)ws_size;
  const float* s      = (const float*)d_in[0];
  const float* v      = (const float*)d_in[1];
  const int*   batch  = (const int*)d_in[2];
  const float* weight = (const float*)d_in[3];
  const float* bias   = (const float*)d_in[4];
  float* out = (float*)d_out;
  const int n = in_sizes[0] / SDIM;   // N = 200000
  seg_norm_kernel<<<NSEG, 256, 0, stream>>>(s, v, batch, weight, bias, out, n);
}